// SelfAttention_17721035063939
// MI455X (gfx1250) — compile-verified
//
#include <hip/hip_runtime.h>
#include <math.h>

typedef __attribute__((ext_vector_type(2))) float v2f;
typedef __attribute__((ext_vector_type(8))) float v8f;

#define S_LEN 4096
#define D_DIM 1024
#define B_DIM 64
#define TILE_PITCH 1028  // 1024 + 4 dword pad -> conflict-free 16-row b64 column reads

// ---------------------------------------------------------------------------
// Async stage: 16 rows x 4KB global -> LDS, one b128 per (thread,row), tracked
// by ASYNCcnt (16 instructions per wave per tile).
// ---------------------------------------------------------------------------
__device__ __forceinline__ void stage_tile_async(const float* __restrict__ gsrc,
                                                 float* ldst, int t)
{
#pragma unroll
  for (int j = 0; j < 16; ++j) {
    const float* g = gsrc + (size_t)j * D_DIM + t * 4;
    unsigned lofs = (unsigned)(size_t)(ldst + j * TILE_PITCH + t * 4);
    asm volatile("global_load_async_to_lds_b128 %0, %1, off"
                 :: "v"(lofs), "v"((unsigned long long)(size_t)g)
                 : "memory");
  }
}

// ---------------------------------------------------------------------------
// Kernel 1: per (batch, S-chunk) fused online-softmax partial.
// Double-buffered async tiles; scores via V_WMMA_F32_16X16X4_F32; context
// accumulated from the same LDS tile (inp read from HBM exactly once).
// ---------------------------------------------------------------------------
__global__ __launch_bounds__(256) void attn_partial_kernel(
    const float* __restrict__ inp, const int* __restrict__ lens,
    const float* __restrict__ w, const float* __restrict__ bias,
    float* __restrict__ cpart, float* __restrict__ mpart,
    float* __restrict__ lpart, int rows_per_chunk)
{
  __shared__ float w_lds[D_DIM];
  __shared__ float tiles[2][16 * TILE_PITCH];
  __shared__ float psc[8 * 16];   // per-wave partial scores
  __shared__ float fsc[16];       // final masked scores for the tile

  const int t    = threadIdx.x;
  const int lane = t & 31;
  const int wv   = t >> 5;
  const int chunk  = blockIdx.x;
  const int bb     = blockIdx.y;
  const int nchunk = gridDim.x;

  // preload w into LDS (each thread: one float4)
  *(float4*)&w_lds[t * 4] = *(const float4*)&w[t * 4];

  const int   len  = lens[bb];
  const float bsc  = bias[0];
  const float* __restrict__ binp   = inp + (size_t)bb * (size_t)(S_LEN * D_DIM);
  const int   row0   = chunk * rows_per_chunk;
  const float* __restrict__ gchunk = binp + (size_t)row0 * D_DIM;
  const int   ntiles = rows_per_chunk >> 4;

  float  m = -__builtin_inff();
  float  l = 0.0f;
  float4 c = make_float4(0.f, 0.f, 0.f, 0.f);

  // prefetch tile 0
  stage_tile_async(gchunk, &tiles[0][0], t);

  for (int ti = 0; ti < ntiles; ++ti) {
    // issue next tile's async loads (writes the buffer everyone finished
    // reading in the previous iteration; end-of-iteration barrier guarantees that)
    if (ti + 1 < ntiles) {
      stage_tile_async(gchunk + (size_t)(ti + 1) * (16 * D_DIM),
                       &tiles[(ti + 1) & 1][0], t);
      asm volatile("s_wait_asynccnt 0x10" ::: "memory");  // current tile landed
    } else {
      asm volatile("s_wait_asynccnt 0x0" ::: "memory");
    }
    __syncthreads();                    // all waves' async writes visible

    float* tile = &tiles[ti & 1][0];

    // ---- scores via WMMA: A = w broadcast (16x4), B = tile^T (4x16) ----
    v8f acc0 = {0.f,0.f,0.f,0.f,0.f,0.f,0.f,0.f};
    v8f acc1 = {0.f,0.f,0.f,0.f,0.f,0.f,0.f,0.f};
    const int koff  = (lane >> 4) * 2;
    const int rowl  = lane & 15;
    const int kbase = wv * 128 + koff;   // each wave owns a K-slice of 128
#pragma unroll 4
    for (int i = 0; i < 16; ++i) {
      const int k0 = kbase + i * 8;
      v2f a0, b0, a1, b1;
      a0.x = w_lds[k0];     a0.y = w_lds[k0 + 1];
      b0.x = tile[rowl * TILE_PITCH + k0];
      b0.y = tile[rowl * TILE_PITCH + k0 + 1];
      acc0 = __builtin_amdgcn_wmma_f32_16x16x4_f32(false, a0, false, b0,
                                                   (short)0, acc0, false, false);
      a1.x = w_lds[k0 + 4]; a1.y = w_lds[k0 + 5];
      b1.x = tile[rowl * TILE_PITCH + k0 + 4];
      b1.y = tile[rowl * TILE_PITCH + k0 + 5];
      acc1 = __builtin_amdgcn_wmma_f32_16x16x4_f32(false, a1, false, b1,
                                                   (short)0, acc1, false, false);
    }
    // D[0][n] sits in vgpr0, lanes 0..15 (all M rows identical by construction)
    if (lane < 16) psc[wv * 16 + lane] = acc0[0] + acc1[0];
    __syncthreads();

    if (t < 16) {
      float s = bsc;
#pragma unroll
      for (int q = 0; q < 8; ++q) s += psc[q * 16 + t];
      fsc[t] = ((row0 + ti * 16 + t) < len) ? s : -__builtin_inff();
    }
    __syncthreads();

    // ---- online softmax update (replicated identically in every thread) ----
    float sv[16];
#pragma unroll
    for (int i = 0; i < 16; ++i) sv[i] = fsc[i];
    float tmax = sv[0];
#pragma unroll
    for (int i = 1; i < 16; ++i) tmax = fmaxf(tmax, sv[i]);
    const float mn = fmaxf(m, tmax);

    float e[16];
    float scale = 1.0f, sume = 0.0f;
    if (mn == -__builtin_inff()) {       // tile fully masked & nothing seen yet
#pragma unroll
      for (int i = 0; i < 16; ++i) e[i] = 0.0f;
    } else {
      scale = __expf(m - mn);            // m == -inf -> 0
#pragma unroll
      for (int i = 0; i < 16; ++i) {
        e[i] = (sv[i] == -__builtin_inff()) ? 0.0f : __expf(sv[i] - mn);
        sume += e[i];
      }
    }
    m = mn;
    l = l * scale + sume;
    c.x *= scale; c.y *= scale; c.z *= scale; c.w *= scale;

    // ---- accumulate context from the same LDS tile (thread t owns 4 cols) ----
#pragma unroll
    for (int i = 0; i < 16; ++i) {
      float4 x = *(float4*)&tile[i * TILE_PITCH + t * 4];
      c.x += e[i] * x.x; c.y += e[i] * x.y;
      c.z += e[i] * x.z; c.w += e[i] * x.w;
    }
    __syncthreads();   // everyone done reading before this buffer is re-staged
  }

  const size_t pb = (size_t)bb * nchunk + chunk;
  *(float4*)&cpart[pb * D_DIM + t * 4] = c;
  if (t == 0) { mpart[pb] = m; lpart[pb] = l; }
}

// ---------------------------------------------------------------------------
// Kernel 2: merge per-chunk partials: out[b,:] = sum_i e^{m_i-M} c_i / L
// ---------------------------------------------------------------------------
__global__ __launch_bounds__(256) void attn_combine_kernel(
    const float* __restrict__ cpart, const float* __restrict__ mpart,
    const float* __restrict__ lpart, float* __restrict__ out, int nchunk)
{
  const int bb = blockIdx.x;
  const int t  = threadIdx.x;
  const size_t base = (size_t)bb * nchunk;

  float M = -__builtin_inff();
  for (int i = 0; i < nchunk; ++i) M = fmaxf(M, mpart[base + i]);

  float  L   = 0.0f;
  float4 acc = make_float4(0.f, 0.f, 0.f, 0.f);
  for (int i = 0; i < nchunk; ++i) {
    const float f = __expf(mpart[base + i] - M);   // empty chunk: m=-inf -> 0
    L += lpart[base + i] * f;
    float4 v = *(const float4*)&cpart[(base + i) * D_DIM + t * 4];
    acc.x += f * v.x; acc.y += f * v.y; acc.z += f * v.z; acc.w += f * v.w;
  }
  const float inv = 1.0f / L;   // L > 0 guaranteed (len >= 1)
  acc.x *= inv; acc.y *= inv; acc.z *= inv; acc.w *= inv;
  *(float4*)&out[(size_t)bb * D_DIM + t * 4] = acc;
}

// ---------------------------------------------------------------------------
extern "C" void kernel_launch(void* const* d_in, const int* in_sizes, int n_in,
                              void* d_out, int out_size, void* d_ws, size_t ws_size,
                              hipStream_t stream)
{
  (void)in_sizes; (void)n_in; (void)out_size;
  const float* inp  = (const float*)d_in[0];
  const int*   lens = (const int*)d_in[1];
  const float* w    = (const float*)d_in[2];
  const float* bias = (const float*)d_in[3];
  float*       out  = (float*)d_out;

  // pick the largest chunk count whose partial buffers fit in the workspace
  int nchunk = 1;
  const int cands[7] = {64, 32, 16, 8, 4, 2, 1};
  for (int i = 0; i < 7; ++i) {
    const size_t need = (size_t)B_DIM * cands[i] * (D_DIM + 2) * sizeof(float);
    if (need <= ws_size) { nchunk = cands[i]; break; }
  }
  const int rows_per_chunk = S_LEN / nchunk;

  float* cpart = (float*)d_ws;
  float* mpart = cpart + (size_t)B_DIM * nchunk * D_DIM;
  float* lpart = mpart + (size_t)B_DIM * nchunk;

  dim3 g1(nchunk, B_DIM);
  attn_partial_kernel<<<g1, 256, 0, stream>>>(inp, lens, w, bias,
                                              cpart, mpart, lpart, rows_per_chunk);
  attn_combine_kernel<<<B_DIM, 256, 0, stream>>>(cpart, mpart, lpart, out, nchunk);
}